// SCAMPLayer_59072980189858
// MI455X (gfx1250) — compile-verified
//
#include <hip/hip_runtime.h>
#include <hip/hip_bf16.h>
#include <math.h>

// Dims from reference
#define Bsz   512
#define Ndim  3072     // n = Mr*Lr
#define LMdim 2048     // Mc*Lc
#define LRdim 48
#define LCdim 32

typedef float v2f __attribute__((ext_vector_type(2)));
typedef float v8f __attribute__((ext_vector_type(8)));

#define WMMA_F32(a, b, c) \
  __builtin_amdgcn_wmma_f32_16x16x4_f32(false, (a), false, (b), (short)0, (c), false, false)

// ---------------------------------------------------------------------------
// Kernel 1: per-batch small solves.
//   gma = (W @ psi)/Lc ; phi_new = sigma2 + gma ; b = gma/phi ;
//   tau = (L / (W^T @ (1/phi_new))) / Mr = 2 / (W^T @ invphi)
// ---------------------------------------------------------------------------
__global__ __launch_bounds__(64) void scamp_prep(
    const float* __restrict__ W,        // (48,32)
    const float* __restrict__ psi,      // (512,32)
    const float* __restrict__ phi,      // (512,48)
    const float* __restrict__ sigma2p,  // scalar
    float* __restrict__ phi_new,        // out (512,48)
    float* __restrict__ bons,           // ws (512,48)
    float* __restrict__ invphi,         // ws (512,48)
    float* __restrict__ tauw)           // ws (512,32)
{
  const int b = blockIdx.x;
  const int tid = threadIdx.x;
  __shared__ float psi_s[LCdim];
  __shared__ float ip_s[LRdim];

  if (tid < LCdim) psi_s[tid] = psi[b * LCdim + tid];
  __syncthreads();

  if (tid < LRdim) {
    float g = 0.0f;
#pragma unroll
    for (int lc = 0; lc < LCdim; ++lc) g += W[tid * LCdim + lc] * psi_s[lc];
    g *= (1.0f / (float)LCdim);
    const float pn = sigma2p[0] + g;
    const float ip = 1.0f / pn;
    phi_new[b * LRdim + tid] = pn;
    invphi[b * LRdim + tid] = ip;
    bons[b * LRdim + tid] = g / phi[b * LRdim + tid];
    ip_s[tid] = ip;
  }
  __syncthreads();

  if (tid < LCdim) {
    float s = 0.0f;
#pragma unroll
    for (int lr = 0; lr < LRdim; ++lr) s += W[lr * LCdim + tid] * ip_s[lr];
    tauw[b * LCdim + tid] = 2.0f / s;   // (128/s)/64
  }
}

// ---------------------------------------------------------------------------
// Kernel 2: z_new = y - A @ xmmse + repeat(b,64) * z
// Complex GEMM: C(512 x 3072) = Xmmse(512 x 2048) * A^T, via f32 WMMA 16x16x4.
// Each wave computes a 32(batch) x 64(i) strip: 2 M-subtiles share every
// B-fragment load -> 8 WMMAs per 16-byte B load. Outer loop of 16 k-units
// hoists branch-free prefetch; inner 4 sub-steps fully unrolled.
// ---------------------------------------------------------------------------
__global__ __launch_bounds__(128) void scamp_gemm1_znew(
    const float* __restrict__ Amat,   // (3072,2048) complex interleaved
    const float* __restrict__ xmmse,  // (512,2048) complex
    const float* __restrict__ y,      // (512,3072) complex
    const float* __restrict__ z,      // (512,3072) complex
    const float* __restrict__ bons,   // (512,48)
    float* __restrict__ znew)         // (512,3072) complex
{
  const int lane = threadIdx.x & 31;
  const int wave = threadIdx.x >> 5;                 // 0..3
  const int b0 = blockIdx.y * 32;                    // batch tile (2 subtiles)
  const int i0 = blockIdx.x * 256 + wave * 64;       // column strip
  const int mr = lane & 15;
  const int kp = (lane >> 4) << 1;                   // 0 or 2

  v8f cre[2][4] = {};
  v8f cim[2][4] = {};

  const float* __restrict__ xbase =
      xmmse + ((size_t)(b0 + mr) * LMdim + kp) * 2;       // + u*16 rows via imm
  const float* __restrict__ bbase =
      Amat + ((size_t)(i0 + mr) * LMdim + kp) * 2;        // + t*16 rows via imm

  for (int kk0 = 0; kk0 < LMdim; kk0 += 16) {
    // branch-free prefetch: one 128B line per row-group, 512B ahead
    __builtin_prefetch(xbase + (size_t)(kk0 + 64) * 2, 0, 3);
    __builtin_prefetch(xbase + (size_t)16 * LMdim * 2 + (size_t)(kk0 + 64) * 2, 0, 3);
#pragma unroll
    for (int t = 0; t < 4; ++t)
      __builtin_prefetch(bbase + (size_t)t * 16 * LMdim * 2 +
                         (size_t)(kk0 + 64) * 2, 0, 3);

#pragma unroll
    for (int s = 0; s < 4; ++s) {
      const int kk = kk0 + s * 4;

      v2f are[2], aim[2], aimn[2];
#pragma unroll
      for (int u = 0; u < 2; ++u) {
        const float4 af = *(const float4*)(xbase + (size_t)u * 16 * LMdim * 2 +
                                           (size_t)kk * 2);
        are[u]  = v2f{af.x, af.z};
        aim[u]  = v2f{af.y, af.w};
        aimn[u] = v2f{-af.y, -af.w};
      }
#pragma unroll
      for (int t = 0; t < 4; ++t) {
        const float4 bf = *(const float4*)(bbase + (size_t)t * 16 * LMdim * 2 +
                                           (size_t)kk * 2);
        const v2f bre = {bf.x, bf.z};
        const v2f bim = {bf.y, bf.w};
#pragma unroll
        for (int u = 0; u < 2; ++u) {
          // c_re += are*bre - aim*bim ; c_im += are*bim + aim*bre
          cre[u][t] = WMMA_F32(aimn[u], bim, cre[u][t]);
          cre[u][t] = WMMA_F32(are[u],  bre, cre[u][t]);
          cim[u][t] = WMMA_F32(aim[u],  bre, cim[u][t]);
          cim[u][t] = WMMA_F32(are[u],  bim, cim[u][t]);
        }
      }
    }
  }

  // Epilogue: z_new = y - C + b*z  (C layout: n = lane&15, m = v + 8*(lane>=16))
  const int mbase = (lane >> 4) << 3;
#pragma unroll
  for (int u = 0; u < 2; ++u) {
#pragma unroll
    for (int t = 0; t < 4; ++t) {
      const int n = i0 + t * 16 + mr;
      const int bi = n >> 6;                         // i / Mr
#pragma unroll
      for (int v = 0; v < 8; ++v) {
        const int bb = b0 + u * 16 + mbase + v;
        const size_t idx = ((size_t)bb * Ndim + n) * 2;
        const float bo = bons[bb * LRdim + bi];
        znew[idx + 0] = y[idx + 0] - cre[u][t][v] + bo * z[idx + 0];
        znew[idx + 1] = y[idx + 1] - cim[u][t][v] + bo * z[idx + 1];
      }
    }
  }
}

// ---------------------------------------------------------------------------
// Kernel 3: xmap = xmmse + tau_use * (conj(A)^T @ (z_new / phi_use))
// Complex GEMM: U(512 x 2048) = Zs(512 x 3072) * conj(A), K = 3072.
// Same 32x64 per-wave strip; B-fragments (rows of A) shared by both M-subtiles.
// ---------------------------------------------------------------------------
__global__ __launch_bounds__(128) void scamp_gemm2_xmap(
    const float* __restrict__ Amat,    // (3072,2048) complex
    const float* __restrict__ znew,    // (512,3072) complex
    const float* __restrict__ invphi,  // (512,48)
    const float* __restrict__ tauw,    // (512,32)
    const float* __restrict__ xmmse,   // (512,2048) complex
    float* __restrict__ xmap)          // (512,2048) complex
{
  const int lane = threadIdx.x & 31;
  const int wave = threadIdx.x >> 5;
  const int b0 = blockIdx.y * 32;
  const int n0 = blockIdx.x * 256 + wave * 64;       // column strip over LM
  const int mr = lane & 15;
  const int kp = (lane >> 4) << 1;

  v8f cre[2][4] = {};
  v8f cim[2][4] = {};

  const float* __restrict__ zbase =
      znew + ((size_t)(b0 + mr) * Ndim + kp) * 2;         // + u*16 rows via imm
  const float* __restrict__ ipr0 = invphi + (b0 + mr) * LRdim;
  const float* __restrict__ ipr1 = ipr0 + 16 * LRdim;
  const float* __restrict__ bbase =
      Amat + ((size_t)kp * LMdim + n0 + mr) * 2;          // rows kk+kp via kk term

  for (int kk0 = 0; kk0 < Ndim; kk0 += 16) {
    // prefetch the z-stream 512B ahead (A-row stream is sequential/shared -> L2)
    __builtin_prefetch(zbase + (size_t)(kk0 + 64) * 2, 0, 3);
    __builtin_prefetch(zbase + (size_t)16 * Ndim * 2 + (size_t)(kk0 + 64) * 2, 0, 3);

#pragma unroll
    for (int s = 0; s < 4; ++s) {
      const int kk = kk0 + s * 4;

      v2f are[2], aim[2], aren[2];
#pragma unroll
      for (int u = 0; u < 2; ++u) {
        const float4 zf = *(const float4*)(zbase + (size_t)u * 16 * Ndim * 2 +
                                           (size_t)kk * 2);
        const float ip = (u ? ipr1 : ipr0)[kk >> 6]; // i/64 const over 4-k step
        are[u]  = v2f{zf.x * ip, zf.z * ip};
        aim[u]  = v2f{zf.y * ip, zf.w * ip};
        aren[u] = v2f{-are[u].x, -are[u].y};
      }
#pragma unroll
      for (int t = 0; t < 4; ++t) {
        const float2 bf0 = *(const float2*)(bbase + (size_t)kk * LMdim * 2 +
                                            (size_t)t * 32);
        const float2 bf1 = *(const float2*)(bbase + (size_t)kk * LMdim * 2 +
                                            (size_t)LMdim * 2 + (size_t)t * 32);
        const v2f bre = {bf0.x, bf1.x};
        const v2f bim = {bf0.y, bf1.y};
#pragma unroll
        for (int u = 0; u < 2; ++u) {
          // conj(A): u_re += are*bre + aim*bim ; u_im += aim*bre - are*bim
          cre[u][t] = WMMA_F32(aim[u],  bim, cre[u][t]);
          cre[u][t] = WMMA_F32(are[u],  bre, cre[u][t]);
          cim[u][t] = WMMA_F32(aim[u],  bre, cim[u][t]);
          cim[u][t] = WMMA_F32(aren[u], bim, cim[u][t]);
        }
      }
    }
  }

  const int mbase = (lane >> 4) << 3;
#pragma unroll
  for (int u = 0; u < 2; ++u) {
#pragma unroll
    for (int t = 0; t < 4; ++t) {
      const int k = n0 + t * 16 + mr;
      const int tc = k >> 6;                         // k / Mc
#pragma unroll
      for (int v = 0; v < 8; ++v) {
        const int bb = b0 + u * 16 + mbase + v;
        const size_t idx = ((size_t)bb * LMdim + k) * 2;
        const float tv = tauw[bb * LCdim + tc];
        xmap[idx + 0] = xmmse[idx + 0] + tv * cre[u][t][v];
        xmap[idx + 1] = xmmse[idx + 1] + tv * cim[u][t][v];
      }
    }
  }
}

// ---------------------------------------------------------------------------
// Kernel 4: QPSK softmax denoiser. One thread per (b, l) group (M=16, K=4).
// The reference's single global max cancels exactly in num/den, so a
// per-(b,l) max is mathematically identical and numerically stable.
// ---------------------------------------------------------------------------
__global__ __launch_bounds__(256) void scamp_denoise(
    const float* __restrict__ xmap,    // (512,2048) complex
    const float* __restrict__ tauw,    // (512,32)
    const float* __restrict__ symc,    // (4,) complex -> 8 floats
    float* __restrict__ xnew)          // (512,2048) complex
{
  const int t = blockIdx.x * blockDim.x + threadIdx.x;   // 65536
  const int b = t >> 7;
  const int l = t & 127;

  const float th = tauw[b * LCdim + (l >> 2)] * 0.5f;    // tau_use/2
  const float it = 1.0f / th;

  const float c0 = symc[0], d0 = symc[1];
  const float c1 = symc[2], d1 = symc[3];
  const float c2 = symc[4], d2 = symc[5];
  const float c3 = symc[6], d3 = symc[7];

  const float* __restrict__ xp = xmap + ((size_t)b * LMdim + l * 16) * 2;

  float xr[16], xi[16];
  float xmax = -3.402823466e38f;
#pragma unroll
  for (int m = 0; m < 16; ++m) {
    const float sr = xp[2 * m] * it;
    const float si = xp[2 * m + 1] * it;
    xr[m] = sr; xi[m] = si;
    const float x0 = sr * c0 + si * d0;
    const float x1 = sr * c1 + si * d1;
    const float x2 = sr * c2 + si * d2;
    const float x3 = sr * c3 + si * d3;
    xmax = fmaxf(xmax, fmaxf(fmaxf(x0, x1), fmaxf(x2, x3)));
  }

  float den = 0.0f;
  float nr[16], ni[16];
#pragma unroll
  for (int m = 0; m < 16; ++m) {
    const float e0 = expf(xr[m] * c0 + xi[m] * d0 - xmax);
    const float e1 = expf(xr[m] * c1 + xi[m] * d1 - xmax);
    const float e2 = expf(xr[m] * c2 + xi[m] * d2 - xmax);
    const float e3 = expf(xr[m] * c3 + xi[m] * d3 - xmax);
    nr[m] = c0 * e0 + c1 * e1 + c2 * e2 + c3 * e3;
    ni[m] = d0 * e0 + d1 * e1 + d2 * e2 + d3 * e3;
    den += e0 + e1 + e2 + e3;
  }
  const float id = 1.0f / den;

  float* __restrict__ op = xnew + ((size_t)b * LMdim + l * 16) * 2;
#pragma unroll
  for (int m = 0; m < 16; ++m) {
    op[2 * m + 0] = nr[m] * id;
    op[2 * m + 1] = ni[m] * id;
  }
}

// ---------------------------------------------------------------------------
// Kernel 5: psi_new = 1 - sum_{Mc} |xmmse_new|^2 / Na, grouped (B, Lc, Mc)
// ---------------------------------------------------------------------------
__global__ __launch_bounds__(256) void scamp_psi(
    const float* __restrict__ xnew,    // (512,2048) complex
    float* __restrict__ psi_new)       // (512,32)
{
  const int t = blockIdx.x * blockDim.x + threadIdx.x;   // 16384
  const int b = t >> 5;
  const int lc = t & 31;
  const float4* __restrict__ p =
      (const float4*)(xnew + ((size_t)b * LMdim + lc * 64) * 2);
  float s = 0.0f;
#pragma unroll
  for (int j = 0; j < 32; ++j) {                         // 64 complex = 32 float4
    const float4 v = p[j];
    s += v.x * v.x + v.y * v.y + v.z * v.z + v.w * v.w;
  }
  psi_new[t] = 1.0f - s * 0.25f;                         // /Na
}

// ---------------------------------------------------------------------------
extern "C" void kernel_launch(void* const* d_in, const int* in_sizes, int n_in,
                              void* d_out, int out_size, void* d_ws, size_t ws_size,
                              hipStream_t stream) {
  const float* W      = (const float*)d_in[0];
  const float* A      = (const float*)d_in[1];
  const float* y      = (const float*)d_in[2];
  const float* z      = (const float*)d_in[3];
  const float* psi    = (const float*)d_in[4];
  const float* phi    = (const float*)d_in[5];
  const float* xmmse  = (const float*)d_in[6];
  const float* sigma2 = (const float*)d_in[7];
  const float* symc   = (const float*)d_in[8];

  float* out = (float*)d_out;
  float* z_new   = out;                                  // 512*3072*2 = 3145728
  float* phi_new = out + 3145728;                        // 512*48     = 24576
  float* xmap    = out + 3145728 + 24576;                // 512*2048*2 = 2097152
  float* xn      = out + 3145728 + 24576 + 2097152;      // 512*2048*2 = 2097152
  float* psi_new = out + 3145728 + 24576 + 2097152 + 2097152;  // 512*32

  float* ws     = (float*)d_ws;
  float* bons   = ws;                                    // 512*48
  float* invphi = ws + 24576;                            // 512*48
  float* tauw   = ws + 49152;                            // 512*32

  scamp_prep<<<Bsz, 64, 0, stream>>>(W, psi, phi, sigma2, phi_new, bons,
                                     invphi, tauw);

  scamp_gemm1_znew<<<dim3(Ndim / 256, Bsz / 32), 128, 0, stream>>>(
      A, xmmse, y, z, bons, z_new);

  scamp_gemm2_xmap<<<dim3(LMdim / 256, Bsz / 32), 128, 0, stream>>>(
      A, z_new, invphi, tauw, xmmse, xmap);

  scamp_denoise<<<(Bsz * 128) / 256, 256, 0, stream>>>(xmap, tauw, symc, xn);

  scamp_psi<<<(Bsz * 32) / 256, 256, 0, stream>>>(xn, psi_new);
}